// HCentroidDistance_2954937500464
// MI455X (gfx1250) — compile-verified
//
#include <hip/hip_runtime.h>
#include <hip/hip_bf16.h>
#include <math.h>

typedef __attribute__((ext_vector_type(2))) float v2f;
typedef __attribute__((ext_vector_type(8))) float v8f;

#define EPS_F 1e-05f
#define MIN_NORM_F 1e-15f
#define HALF_LN2_F 0.34657359027997264f   // 0.5 * ln(2)

constexpr int Bn = 8, Tn = 12, Nn = 325, Cn = 32, Kn = 100;
constexpr int M_TOTAL = Bn * Tn * Nn;      // 31200 = 1950 * 16 exactly
constexpr int M_TILES = M_TOTAL / 16;      // 1950
constexpr int K_TILES = 7;                 // 112 padded centroids
constexpr int KPAD = K_TILES * 16;         // 112

// ---------------------------------------------------------------------------
// Kernel 1: centroids = expmap0(centroid_weight, c);  also y2 = ||centroid||^2
// Padded centroids (k >= 100) are zeroed so the WMMA B-operand is safe.
// ---------------------------------------------------------------------------
__global__ void prep_centroids_kernel(const float* __restrict__ w,
                                      const float* __restrict__ cptr,
                                      float* __restrict__ cent,
                                      float* __restrict__ y2out) {
    int k = blockIdx.x * blockDim.x + threadIdx.x;
    if (k >= KPAD) return;
    float c  = cptr[0];
    float sc = sqrtf(c);
    if (k < Kn) {
        float u[Cn];
        float s = 0.f;
#pragma unroll
        for (int j = 0; j < Cn; ++j) { u[j] = w[k * Cn + j]; s = fmaf(u[j], u[j], s); }
        float un    = fmaxf(sqrtf(s), MIN_NORM_F);
        float t     = tanhf(sc * un);
        float scale = t / (sc * un);
#pragma unroll
        for (int j = 0; j < Cn; ++j) cent[k * Cn + j] = scale * u[j];
        float ny = t / sc;            // ||y|| = tanh(sc*||u||)/sc
        y2out[k] = ny * ny;
    } else {
#pragma unroll
        for (int j = 0; j < Cn; ++j) cent[k * Cn + j] = 0.f;
        y2out[k] = 0.f;
    }
}

// ---------------------------------------------------------------------------
// Kernel 2: fused xy-GEMM (V_WMMA_F32_16X16X4_F32, K=32 via 8 chained WMMAs)
// + hyperbolic sqdist epilogue using HW rcp/sqrt/log2 (v_rcp_f32, v_sqrt_f32,
// v_log_f32), ~1 ulp each. One wave (32 lanes) per 16-row output tile.
// ---------------------------------------------------------------------------
__global__ void __launch_bounds__(32)
hdist_main_kernel(const float* __restrict__ x,       // [31200][32]
                  const float* __restrict__ cent,    // [112][32]  (padded)
                  const float* __restrict__ y2arr,   // [112]
                  const float* __restrict__ cptr,
                  float* __restrict__ nodeDist) {    // [31200][100]
    const int tile  = blockIdx.x;          // 0..1949
    const int lane  = threadIdx.x;         // 0..31
    const int mbase = tile * 16;
    const int l16   = lane & 15;
    const int lhalf = lane >> 4;           // 0: K-pair {0,1}, 1: K-pair {2,3}

    const float c  = cptr[0];
    const float sc = __builtin_amdgcn_sqrtf(c);

    // --- Preload A operand: row (mbase+l16), 8 chunks of 4 K-values split
    //     across lane halves per the 32-bit 16x4 A layout.
    v2f a[8];
    float part = 0.f;
    const float* xrow = x + (size_t)(mbase + l16) * Cn + lhalf * 2;
#pragma unroll
    for (int j = 0; j < 8; ++j) {
        a[j] = *(const v2f*)(xrow + j * 4);
        part = fmaf(a[j].x, a[j].x, fmaf(a[j].y, a[j].y, part));
    }
    // lane L and L+16 each hold half of row (L&15)'s ||x||^2; combine.
    float x2full = part + __shfl_xor(part, 16, 32);  // x2 of row l16 (both halves)

    // --- Hoist per-row broadcasts out of the centroid-tile loop:
    //     acc[r] maps to row (r + lhalf*8); broadcast that row's x2 once.
    float x2r[8], Br[8], B2r[8], ccx2r[8];
#pragma unroll
    for (int r = 0; r < 8; ++r) {
        float x2 = __shfl(x2full, r + lhalf * 8, 32);
        x2r[r]   = x2;
        Br[r]    = 1.f - c * x2;       // B = 1 - c*x2   (per-row constant)
        B2r[r]   = Br[r] * Br[r];
        ccx2r[r] = c * c * x2;
    }

    const float inv2sc = 2.f / sc;

    for (int kt = 0; kt < K_TILES; ++kt) {
        // --- B operand: centroid column (kt*16 + l16), same half-lane K split.
        const float* crow = cent + (size_t)(kt * 16 + l16) * Cn + lhalf * 2;
        v8f acc = {};
#pragma unroll
        for (int j = 0; j < 8; ++j) {
            v2f b = *(const v2f*)(crow + j * 4);
            acc = __builtin_amdgcn_wmma_f32_16x16x4_f32(
                /*neg_a=*/false, a[j], /*neg_b=*/false, b,
                /*c_mod=*/(short)0, acc, /*reuse_a=*/false, /*reuse_b=*/false);
        }

        const int   kcol = kt * 16 + l16;     // centroid index for this lane
        const float y2   = y2arr[kcol];
        const float cy2  = c * y2;

        // C/D layout: acc[r] -> row (r + lhalf*8), col l16
#pragma unroll
        for (int r = 0; r < 8; ++r) {
            const float xy   = acc[r];
            const float base = fmaf(-2.f * c, xy, 1.f);           // 1 - 2c*xy
            const float A_   = base + cy2;                        // 1-2c*xy+c*y2
            const float D_   = fmaxf(fmaf(ccx2r[r], y2, base), MIN_NORM_F);

            // ||num||^2 = A^2*x2 - 2AB*xy + B^2*y2
            float n2 = fmaf(A_ * A_, x2r[r],
                            fmaf(-2.f * A_ * Br[r], xy, B2r[r] * y2));
            float nrm = __builtin_amdgcn_sqrtf(fmaxf(n2, 0.f)) *
                        __builtin_amdgcn_rcpf(D_);

            float z = sc * nrm;
            z = fminf(fmaxf(z, -1.f + EPS_F), 1.f - EPS_F);
            // artanh(z) = 0.5*ln2*log2((1+z)/(1-z)) via v_log_f32 + v_rcp_f32
            float ratio = (1.f + z) * __builtin_amdgcn_rcpf(1.f - z);
            float at    = HALF_LN2_F * __builtin_amdgcn_logf(ratio);
            float d     = inv2sc * at;
            d = d * d;

            if (kcol < Kn) {
                const size_t m = (size_t)(mbase + r + lhalf * 8);
                nodeDist[m * Kn + kcol] = d;
            }
        }
    }
}

// ---------------------------------------------------------------------------
// Kernel 3: graph_centroid_dist[b,n,k] = sum_t node_dist[b,t,n,k] / N
// Deterministic (no atomics); reads are coalesced over k.
// ---------------------------------------------------------------------------
__global__ void graph_reduce_kernel(const float* __restrict__ nodeDist,
                                    float* __restrict__ graph) {
    int i = blockIdx.x * blockDim.x + threadIdx.x;   // over B*N*K
    if (i >= Bn * Nn * Kn) return;
    int k = i % Kn;
    int n = (i / Kn) % Nn;
    int b = i / (Kn * Nn);
    float s = 0.f;
#pragma unroll
    for (int t = 0; t < Tn; ++t)
        s += nodeDist[(size_t)((b * Tn + t) * Nn + n) * Kn + k];
    graph[i] = s * (1.0f / (float)Nn);
}

// ---------------------------------------------------------------------------
extern "C" void kernel_launch(void* const* d_in, const int* in_sizes, int n_in,
                              void* d_out, int out_size, void* d_ws, size_t ws_size,
                              hipStream_t stream) {
    const float* node_repr = (const float*)d_in[0];   // [8,12,325,32]
    const float* cweight   = (const float*)d_in[1];   // [100,32]
    const float* cvec      = (const float*)d_in[2];   // [1]

    float* graph    = (float*)d_out;                  // [8,325,100]  = 260000
    float* nodeDist = graph + (size_t)Bn * Nn * Kn;   // [8,12,325,100] = 3120000

    float* cent = (float*)d_ws;                       // [112][32]
    float* y2   = cent + (size_t)KPAD * Cn;           // [112]

    prep_centroids_kernel<<<1, 128, 0, stream>>>(cweight, cvec, cent, y2);

    hdist_main_kernel<<<M_TILES, 32, 0, stream>>>(node_repr, cent, y2, cvec, nodeDist);

    const int gElems = Bn * Nn * Kn;
    graph_reduce_kernel<<<(gElems + 255) / 256, 256, 0, stream>>>(nodeDist, graph);
}